// DGLBatchCapsuleLayer_48172353192513
// MI455X (gfx1250) — compile-verified
//
#include <hip/hip_runtime.h>
#include <hip/hip_bf16.h>

typedef __attribute__((ext_vector_type(2))) float v2f;
typedef __attribute__((ext_vector_type(4))) float v4f;
typedef __attribute__((ext_vector_type(8))) float v8f;

#define CDIM 1152
#define JDIM 10
#define SDIM 16
#define IDIM 8
#define BDIM 512
#define KDIM (CDIM * IDIM) /* 9216 */

// ---------------------------------------------------------------- init b_ij
__global__ void k_init_b(float* __restrict__ b_ij) {
  int t = blockIdx.x * blockDim.x + threadIdx.x; // grid sized exactly 11520
  b_ij[t] = 0.0f;
}

// ------------------------------------------------- softmax over J per row c
__global__ void k_softmax(const float* __restrict__ b_ij,
                          float* __restrict__ coeff) {
  int c = blockIdx.x * blockDim.x + threadIdx.x;
  if (c >= CDIM) return;
  float row[JDIM];
  float m = -3.4e38f;
#pragma unroll
  for (int j = 0; j < JDIM; ++j) { row[j] = b_ij[c * JDIM + j]; m = fmaxf(m, row[j]); }
  float sum = 0.0f;
#pragma unroll
  for (int j = 0; j < JDIM; ++j) { row[j] = __expf(row[j] - m); sum += row[j]; }
  float inv = 1.0f / sum;
#pragma unroll
  for (int j = 0; j < JDIM; ++j) coeff[c * JDIM + j] = row[j] * inv;
}

// --------------------------------------------------------------- s-GEMM
// s[j,b,s16] = sum_{k=(c,i)} (coeff[c,j]*W[c,j,s16,i]) * x[b,i,c]
// M = 16 (s16, tile == one j), N = 16 b's per block, K = 9216 split over 8 waves.
__global__ void __launch_bounds__(256) k_s_gemm(const float* __restrict__ x,
                                                const float* __restrict__ w,
                                                const float* __restrict__ coeff,
                                                float* __restrict__ s_out) {
  const int nt   = blockIdx.x;          // 0..31  : b-tile
  const int j    = blockIdx.y;          // 0..9   : m-tile == j
  const int lane = threadIdx.x & 31;
  const int wv   = threadIdx.x >> 5;    // 0..7
  const int hi   = lane >> 4;           // half-wave select
  const int l15  = lane & 15;
  const int n    = nt * 16 + l15;       // global b column

  __shared__ float red[8 * 256];

  v8f acc = {};
  const int kbase = wv * (KDIM / 8);    // 1152 per wave
#pragma unroll 2
  for (int kc = 0; kc < (KDIM / 8) / 4; ++kc) {  // 288 WMMAs per wave
    int   k0 = kbase + kc * 4;
    int   c  = k0 >> 3;                 // k = c*8 + i ; 4 k's share one c
    int   i0 = k0 & 7;                  // 0 or 4
    float cf = coeff[c * JDIM + j];     // wave-uniform
    // A tile: W[c,j,m=l15,i0..i0+3] * cf   (16B-aligned float4)
    const v4f wq = *(const v4f*)&w[((c * JDIM + j) * SDIM + l15) * IDIM + i0];
    float a0 = hi ? wq.z : wq.x;
    float a1 = hi ? wq.w : wq.y;
    v2f  a = { a0 * cf, a1 * cf };
    // B tile: x[(n*8 + i)*1152 + c], K rows r and r+1 for this half-wave
    int   i  = i0 + (hi << 1);
    const float* xp = &x[(n * IDIM + i) * CDIM + c];
    v2f  b = { xp[0], xp[CDIM] };
    acc = __builtin_amdgcn_wmma_f32_16x16x4_f32(false, a, false, b,
                                                (short)0, acc, false, false);
  }
  // split-K reduction across the 8 waves via LDS (identical D layouts)
  *(v8f*)&red[wv * 256 + lane * 8] = acc;
  __syncthreads();
  const int t = threadIdx.x;
  float sum = 0.0f;
#pragma unroll
  for (int q = 0; q < 8; ++q) sum += red[q * 256 + t];
  int lane_ = t >> 3, r = t & 7;
  int m  = r + ((lane_ >> 4) << 3);     // s16
  int nn = lane_ & 15;                  // b within tile
  s_out[(j * BDIM + (nt * 16 + nn)) * SDIM + m] = sum;
}

// ------------------------------------------------------------- squash + out
__global__ void k_squash(const float* __restrict__ s_in,
                         float* __restrict__ v_out,
                         float* __restrict__ out) {
  int idx = blockIdx.x * blockDim.x + threadIdx.x; // == j*512 + b, 5120 total
  int j = idx / BDIM, b = idx % BDIM;
  const v4f* sp = (const v4f*)&s_in[idx * SDIM];
  v4f sv[4] = { sp[0], sp[1], sp[2], sp[3] };
  float msq = 0.0f;
#pragma unroll
  for (int q = 0; q < 4; ++q)
#pragma unroll
    for (int e = 0; e < 4; ++e) msq += sv[q][e] * sv[q][e];
  float mag   = sqrtf(msq);
  float scale = mag / (1.0f + msq);     // mag_sq/(1+mag_sq) * s/mag
  v4f* vp = (v4f*)&v_out[idx * SDIM];
  v4f* op = (v4f*)&out[(b * JDIM + j) * SDIM];
#pragma unroll
  for (int q = 0; q < 4; ++q) {
    v4f r = { sv[q][0]*scale, sv[q][1]*scale, sv[q][2]*scale, sv[q][3]*scale };
    vp[q] = r;
    op[q] = r;
  }
}

// -------------------------------------------------------------- b_ij update
// Z[j][s, (c,i)] = sum_b v[j,b,s]*x[b,i,c]  (WMMA, K=512)
// b_ij[c,j] += (1/B) sum_{s,i} W[c,j,s,i] * Z[j][s,(c,i)]
__global__ void __launch_bounds__(256) k_b_update(const float* __restrict__ x,
                                                  const float* __restrict__ w,
                                                  const float* __restrict__ v_in,
                                                  float* __restrict__ b_ij) {
  const int lane = threadIdx.x & 31;
  const int wv   = threadIdx.x >> 5;
  const int task = blockIdx.x * 8 + wv;   // 0..5759
  const int j    = task / 576;
  const int nt2  = task % 576;            // n-tile: 16 (c,i) cols = 2 c values
  const int hi   = lane >> 4;
  const int l15  = lane & 15;
  const int cbase = nt2 * 2;
  const int c = cbase + ((lane >> 3) & 1);
  const int i = lane & 7;

  v8f acc = {};
#pragma unroll 2
  for (int kc = 0; kc < BDIM / 4; ++kc) { // 128 WMMAs
    int k0 = kc * 4 + (hi << 1);          // k = b
    // A[m=l15, k] = v[j][k][l15]  (stride SDIM floats over b)
    const float* vp = &v_in[(j * BDIM + k0) * SDIM + l15];
    v2f a = { vp[0], vp[SDIM] };
    // B[k, n=(c,i)] = x[(k*8 + i)*1152 + c]  (stride IDIM*CDIM over b)
    const float* xp = &x[(k0 * IDIM + i) * CDIM + c];
    v2f b = { xp[0], xp[IDIM * CDIM] };
    acc = __builtin_amdgcn_wmma_f32_16x16x4_f32(false, a, false, b,
                                                (short)0, acc, false, false);
  }
  // per-lane dot with W[c,j,m,i]; lane holds n=(c,i), rows m = mbase+r
  const int   mbase = hi << 3;
  const float* wp = &w[((c * JDIM + j) * SDIM + mbase) * IDIM + i];
  float partial = 0.0f;
#pragma unroll
  for (int r = 0; r < 8; ++r) partial += acc[r] * wp[r * IDIM];
  // reduce the two c-groups (lanes {0-7,16-23} -> c0, {8-15,24-31} -> c1)
  float p0 = ((lane >> 3) & 1) ? 0.0f : partial;
  float p1 = partial - p0;
#pragma unroll
  for (int off = 16; off >= 1; off >>= 1) {
    p0 += __shfl_xor(p0, off, 32);
    p1 += __shfl_xor(p1, off, 32);
  }
  if (lane == 0) {
    b_ij[cbase * JDIM + j]       += p0 * (1.0f / BDIM);
    b_ij[(cbase + 1) * JDIM + j] += p1 * (1.0f / BDIM);
  }
}

// ------------------------------------------------------------------- driver
extern "C" void kernel_launch(void* const* d_in, const int* in_sizes, int n_in,
                              void* d_out, int out_size, void* d_ws, size_t ws_size,
                              hipStream_t stream) {
  const float* x = (const float*)d_in[0];   // [512, 8, 1152]
  const float* w = (const float*)d_in[1];   // [1152, 10, 16, 8]
  float* out = (float*)d_out;               // [512, 10, 16, 1]
  float* ws  = (float*)d_ws;

  float* b_ij  = ws;                        // 11520
  float* coeff = ws + 11520;                // 11520
  float* s_buf = ws + 23040;                // 81920
  float* v_buf = ws + 23040 + 81920;        // 81920   (total < 1 MB)

  k_init_b<<<45, 256, 0, stream>>>(b_ij);   // 45*256 == 11520
  for (int it = 0; it < 3; ++it) {
    k_softmax<<<(CDIM + 255) / 256, 256, 0, stream>>>(b_ij, coeff);
    k_s_gemm<<<dim3(32, JDIM), 256, 0, stream>>>(x, w, coeff, s_buf);
    k_squash<<<(JDIM * BDIM) / 256, 256, 0, stream>>>(s_buf, v_buf, out);
    if (it < 2)
      k_b_update<<<720, 256, 0, stream>>>(x, w, v_buf, b_ij);
  }
}